// RACNN_54872502174242
// MI455X (gfx1250) — compile-verified
//
#include <hip/hip_runtime.h>
#include <hip/hip_bf16.h>

#define IN_SIZE  448
#define OUT_SIZE 224

#define GLOBAL_AS __attribute__((address_space(1)))
#define LDS_AS    __attribute__((address_space(3)))

typedef int v4i __attribute__((vector_size(16)));

__device__ __forceinline__ GLOBAL_AS v4i* to_global_v4i(const void* p) {
  return (GLOBAL_AS v4i*)(unsigned long long)p;
}
__device__ __forceinline__ LDS_AS v4i* to_lds_v4i(void* p) {
  return (LDS_AS v4i*)(unsigned long long)p;
}

__device__ __forceinline__ float sig10(float z) {
  return 1.0f / (1.0f + __expf(-10.0f * z));
}

// One block per (b, ch, output row jr); 224 threads = 7 wave32, one output col each.
__global__ __launch_bounds__(OUT_SIZE) void racnn_crop_resize_kernel(
    const float* __restrict__ images,  // (B,3,448,448) f32
    const float* __restrict__ locs,    // (B,3) f32
    float* __restrict__ out) {         // (B,3,224,224) f32
  const int jr = blockIdx.x;
  const int ch = blockIdx.y;
  const int b  = blockIdx.z;
  const int jc = threadIdx.x;

  // ---- wave-uniform box parameters from locs[b] (scalar-cached loads) ----
  float tx = locs[3 * b + 0];
  float ty = locs[3 * b + 1];
  float tl = locs[3 * b + 2];
  tl = fmaxf(tl, (float)IN_SIZE / 3.0f);
  tx = fminf(fmaxf(tx, tl), (float)IN_SIZE - tl);
  ty = fminf(fmaxf(ty, tl), (float)IN_SIZE - tl);
  const float w_off = fmaxf(floorf(tx - tl), 0.0f);
  const float h_off = fmaxf(floorf(ty - tl), 0.0f);
  const float w_end = fminf(floorf(tx + tl), (float)IN_SIZE);
  const float h_end = fminf(floorf(ty + tl), (float)IN_SIZE);

  // ---- row sample (uniform across the block) ----
  const float src_r = w_off + (float)jr * (w_end - w_off - 1.0f) / (float)(OUT_SIZE - 1);
  const float r0f = fminf(fmaxf(floorf(src_r), 0.0f), (float)(IN_SIZE - 1));
  const float wr  = src_r - r0f;
  const int   r0  = (int)r0f;
  const int   r1  = min(r0 + 1, IN_SIZE - 1);
  const float r1f = (float)r1;
  const float mr0 = sig10(r0f - w_off) - sig10(r0f - w_end);
  const float mr1 = sig10(r1f - w_off) - sig10(r1f - w_end);

  // ---- column sample (per thread) ----
  const float src_c = h_off + (float)jc * (h_end - h_off - 1.0f) / (float)(OUT_SIZE - 1);
  const float c0f = fminf(fmaxf(floorf(src_c), 0.0f), (float)(IN_SIZE - 1));
  const float wc  = src_c - c0f;
  const int   c0  = (int)c0f;
  const int   c1  = min(c0 + 1, IN_SIZE - 1);
  const float c1f = (float)c1;
  const float mc0 = sig10(c0f - h_off) - sig10(c0f - h_end);
  const float mc1 = sig10(c1f - h_off) - sig10(c1f - h_end);

  // ---- stage image rows r0 and r1 (448 f32 each) into LDS ----
  __shared__ __align__(16) float rows[2 * IN_SIZE];  // [0..447]=r0, [448..895]=r1
  const size_t chan_base = ((size_t)(b * 3 + ch)) * IN_SIZE * IN_SIZE;
  const float* __restrict__ g0 = images + chan_base + (size_t)r0 * IN_SIZE;
  const float* __restrict__ g1 = images + chan_base + (size_t)r1 * IN_SIZE;

#if defined(__gfx1250__) && __has_builtin(__builtin_amdgcn_global_load_async_to_lds_b128)
  {
    // 224 lanes x 16B = 3584B = exactly two 448-float rows. Lanes 0..111 move
    // row r0, lanes 112..223 move row r1. All 7 waves fully active (EXEC=~0).
    const int t = jc;
    const float* gsrc = (t < 112) ? (g0 + t * 4) : (g1 + (t - 112) * 4);
    float* lp = rows + t * 4;
    __builtin_amdgcn_global_load_async_to_lds_b128(to_global_v4i((const void*)gsrc),
                                                   to_lds_v4i((void*)lp),
                                                   /*offset=*/0, /*cpol=*/0);
  }
#if __has_builtin(__builtin_amdgcn_s_wait_asynccnt)
  __builtin_amdgcn_s_wait_asynccnt(0);
#elif defined(__HIP_DEVICE_COMPILE__)
  asm volatile("s_wait_asynccnt 0" ::: "memory");
#endif
#else
  // Fallback: plain VGPR-through copy (coalesced b32 loads + ds stores).
  for (int k = jc; k < IN_SIZE; k += OUT_SIZE) {
    rows[k]           = g0[k];
    rows[IN_SIZE + k] = g1[k];
  }
#endif
  __syncthreads();

  // ---- masked bilinear blend; LDS gathers have stride <= 2 (conflict-free) ----
  const float one_wc = 1.0f - wc;
  const float top = one_wc * (rows[c0] * mc0)           + wc * (rows[c1] * mc1);
  const float bot = one_wc * (rows[IN_SIZE + c0] * mc0) + wc * (rows[IN_SIZE + c1] * mc1);
  const float res = (1.0f - wr) * mr0 * top + wr * mr1 * bot;

  out[((size_t)(b * 3 + ch) * OUT_SIZE + jr) * OUT_SIZE + jc] = res;
}

extern "C" void kernel_launch(void* const* d_in, const int* in_sizes, int n_in,
                              void* d_out, int out_size, void* d_ws, size_t ws_size,
                              hipStream_t stream) {
  const float* images = (const float*)d_in[0];  // (B,3,448,448) f32
  const float* locs   = (const float*)d_in[1];  // (B,3) f32
  float* out = (float*)d_out;                   // (B,3,224,224) f32

  const int B = in_sizes[0] / (3 * IN_SIZE * IN_SIZE);
  dim3 grid(OUT_SIZE, 3, B);
  dim3 block(OUT_SIZE);
  racnn_crop_resize_kernel<<<grid, block, 0, stream>>>(images, locs, out);
}